// FeatureAlign_V2_65738769433167
// MI455X (gfx1250) — compile-verified
//
#include <hip/hip_runtime.h>
#include <hip/hip_bf16.h>
#include <math.h>

// ---------------- constants ----------------
#define BB    4
#define CIN   288
#define COUT  144
#define HH    96
#define WW    96
#define HWp   (HH*WW)          // 9216
#define HS    48
#define WS    48
#define GG    16
#define KKt   9                // 3x3 taps
#define CGg   9                // channels per group
#define OFFC  288              // 2*9*16
#define KDC   1296             // COUT * 9 (deform im2col depth)
#define EPSBN 1e-5f

typedef __attribute__((ext_vector_type(2))) float v2f;
typedef __attribute__((ext_vector_type(8))) float v8f;
typedef unsigned int u32x4 __attribute__((ext_vector_type(4)));
typedef int i32x4 __attribute__((ext_vector_type(4)));
typedef int i32x8 __attribute__((ext_vector_type(8)));

__device__ __forceinline__ v8f wmma_f32x4(v2f a, v2f b, v8f c) {
  // D = A(16x4, f32) * B(4x16, f32) + C(16x16, f32)
  return __builtin_amdgcn_wmma_f32_16x16x4_f32(false, a, false, b, (short)0, c, false, false);
}

// ---- Tensor Data Mover: 2D tile (rows x cols f32, row stride strideElems) -> LDS ----
__device__ __forceinline__ void tdm_load_2d(unsigned ldsOff, const float* gsrc,
                                            unsigned cols, unsigned rows, unsigned strideElems) {
  unsigned long long ga = (unsigned long long)(size_t)gsrc;
  u32x4 g0;
  g0[0] = 1u;                                        // count=1, user descriptor
  g0[1] = ldsOff;                                    // lds_addr (bytes)
  g0[2] = (unsigned)(ga & 0xffffffffull);            // global_addr[31:0]
  g0[3] = (unsigned)((ga >> 32) & 0x01ffffffull) | (2u << 30);  // addr[56:32] | type=2
  i32x8 g1;
  g1[0] = 0x20000;                                   // data_size = 4B
  g1[1] = (int)((cols & 0xffffu) << 16);             // tensor_dim0[15:0]
  g1[2] = (int)((cols >> 16) | ((rows & 0xffffu) << 16));   // td0 hi | td1 lo
  g1[3] = (int)((rows >> 16) | ((cols & 0xffffu) << 16));   // td1 hi | tile_dim0
  g1[4] = (int)(rows & 0xffffu);                     // tile_dim1 (tile_dim2 = 0)
  g1[5] = (int)strideElems;                          // tensor_dim0_stride[31:0]
  g1[6] = 0;
  g1[7] = 0;
  i32x4 z4 = {0, 0, 0, 0};
#if __clang_major__ >= 23
  i32x8 z8 = {0, 0, 0, 0, 0, 0, 0, 0};
  __builtin_amdgcn_tensor_load_to_lds(g0, g1, z4, z4, z8, 0);
#else
  __builtin_amdgcn_tensor_load_to_lds(g0, g1, z4, z4, 0);
#endif
}

__device__ __forceinline__ unsigned lds_off_of(const void* p) {
  return (unsigned)(unsigned long long)(size_t)p;
}

// ---------------- kernel 1: global average pool over HxW ----------------
__global__ void k_gap(const float* __restrict__ feat_l, float* __restrict__ gap) {
  const int bc = blockIdx.x;
  const float* src = feat_l + (size_t)bc * HWp;
  float s = 0.f;
  for (int i = threadIdx.x; i < HWp / 4; i += blockDim.x) {
    float4 v = ((const float4*)src)[i];
    s += v.x + v.y + v.z + v.w;
  }
  __shared__ float red[256];
  red[threadIdx.x] = s;
  __syncthreads();
  for (int off = 128; off > 0; off >>= 1) {
    if ((int)threadIdx.x < off) red[threadIdx.x] += red[threadIdx.x + off];
    __syncthreads();
  }
  if (threadIdx.x == 0) gap[bc] = red[0] * (1.0f / (float)HWp);
}

// ---------------- kernel 2: attention matvec + batch-BN + sigmoid ----------------
__global__ void k_atten(const float* __restrict__ gap, const float* __restrict__ w,
                        const float* __restrict__ gam, const float* __restrict__ bet,
                        float* __restrict__ atten) {
  const int o = blockIdx.x;
  const int t = threadIdx.x;                       // 128 threads
  float acc0 = 0.f, acc1 = 0.f, acc2 = 0.f, acc3 = 0.f;
  for (int c = t; c < CIN; c += 128) {
    float wv = w[o * CIN + c];
    acc0 += wv * gap[0 * CIN + c];
    acc1 += wv * gap[1 * CIN + c];
    acc2 += wv * gap[2 * CIN + c];
    acc3 += wv * gap[3 * CIN + c];
  }
  __shared__ float red[4][128];
  red[0][t] = acc0; red[1][t] = acc1; red[2][t] = acc2; red[3][t] = acc3;
  __syncthreads();
  for (int off = 64; off > 0; off >>= 1) {
    if (t < off)
      for (int b = 0; b < 4; ++b) red[b][t] += red[b][t + off];
    __syncthreads();
  }
  if (t == 0) {
    float y[4]; float m = 0.f;
    for (int b = 0; b < 4; ++b) { y[b] = red[b][0]; m += y[b]; }
    m *= 0.25f;
    float v = 0.f;
    for (int b = 0; b < 4; ++b) { float d = y[b] - m; v += d * d; }
    v *= 0.25f;
    float rs = rsqrtf(v + EPSBN);
    float g = gam[o], be = bet[o];
    for (int b = 0; b < 4; ++b) {
      float xn = (y[b] - m) * rs * g + be;
      atten[b * CIN + o] = 1.0f / (1.0f + __expf(-xn));
    }
  }
}

// ---------------- kernel 3: bilinear 2x upsample; writes planar + group-interleaved ----------------
__global__ void k_upsample(const float* __restrict__ fs, float* __restrict__ fu,
                           float* __restrict__ fup2) {
  const int idx = blockIdx.x * blockDim.x + threadIdx.x;
  if (idx >= BB * COUT * HWp) return;
  const int x = idx % WW;
  const int y = (idx / WW) % HH;
  const int bc = idx / HWp;
  const int b = bc / COUT, c = bc % COUT;
  float sy = (y + 0.5f) * 0.5f - 0.5f;
  float sx = (x + 0.5f) * 0.5f - 0.5f;
  int y0 = (int)floorf(sy); float wy = sy - (float)y0;
  int x0 = (int)floorf(sx); float wx = sx - (float)x0;
  int y0c = min(max(y0, 0), HS - 1), y1c = min(max(y0 + 1, 0), HS - 1);
  int x0c = min(max(x0, 0), WS - 1), x1c = min(max(x0 + 1, 0), WS - 1);
  const float* s = fs + (size_t)bc * HS * WS;
  float v = (1.f - wy) * ((1.f - wx) * s[y0c * WS + x0c] + wx * s[y0c * WS + x1c])
          +         wy * ((1.f - wx) * s[y1c * WS + x0c] + wx * s[y1c * WS + x1c]);
  fu[idx] = v;
  const int g = c / CGg, cl = c % CGg;
  fup2[((size_t)(b * GG + g) * HWp + (y * WW + x)) * 16 + cl] = v;
}

// ---------------- kernel 4: WMMA GEMM feat_arm, TDM double-buffered B ----------------
// grid: (72, 9, 4), block 256 (8 waves); scale (1+atten) folded into A tile
__global__ void k_gemm_arm(const float* __restrict__ feat_l, const float* __restrict__ wconv,
                           const float* __restrict__ atten, float* __restrict__ feat_arm) {
  __shared__ float As[16 * CIN];          // 18 KB
  __shared__ float Bs[2 * 32 * 128];      // 32 KB double buffer
  const int nchunk = blockIdx.x, mt = blockIdx.y, b = blockIdx.z;
  const int tid = threadIdx.x;
  const int p0 = nchunk * 128;
  for (int i = tid; i < 16 * CIN; i += 256) {
    int r = i / CIN, c = i % CIN;
    As[i] = wconv[(size_t)(mt * 16 + r) * CIN + c] * (1.0f + atten[b * CIN + c]);
  }
  if (tid < 32) {   // prologue: DMA chunk 0 into buffer 0
    tdm_load_2d(lds_off_of(Bs), feat_l + (size_t)(b * CIN) * HWp + p0, 128, 32, HWp);
  }
  const int lane = tid & 31;
  const int wv = tid >> 5;
  const int n = lane & 15;
  const int hi = lane >> 4;
  const int ncol = wv * 16 + n;
  v8f acc = {};
  for (int kc = 0; kc < CIN / 32; ++kc) {
    if (tid < 32) {
      if (kc + 1 < CIN / 32) {
        tdm_load_2d(lds_off_of(Bs) + (unsigned)(((kc + 1) & 1) * 32 * 128 * 4),
                    feat_l + (size_t)(b * CIN + (kc + 1) * 32) * HWp + p0, 128, 32, HWp);
        __builtin_amdgcn_s_wait_tensorcnt(1);
      } else {
        __builtin_amdgcn_s_wait_tensorcnt(0);
      }
    }
    __syncthreads();
    const float* Bc = Bs + (kc & 1) * 32 * 128;
    #pragma unroll
    for (int kk = 0; kk < 8; ++kk) {
      int kb = kk * 4 + hi * 2;
      v2f a = *(const v2f*)(As + n * CIN + kc * 32 + kb);
      v2f bb; bb.x = Bc[kb * 128 + ncol]; bb.y = Bc[(kb + 1) * 128 + ncol];
      acc = wmma_f32x4(a, bb, acc);
    }
    __syncthreads();
  }
  const int col = p0 + ncol;
  #pragma unroll
  for (int r = 0; r < 8; ++r) {
    int m = mt * 16 + r + hi * 8;
    feat_arm[(size_t)(b * COUT + m) * HWp + col] = acc[r];
  }
}

// ---------------- kernel 5: WMMA GEMM offsets, TDM double-buffered, transposed store ----------------
// cat scale (x2 for upsample half) folded into A.  grid: (72, 18, 4), block 256
__global__ void k_gemm_off(const float* __restrict__ feat_arm, const float* __restrict__ feat_up,
                           const float* __restrict__ woff, float* __restrict__ offT) {
  __shared__ float As[16 * CIN];
  __shared__ float Bs[2 * 32 * 128];
  const int nchunk = blockIdx.x, mt = blockIdx.y, b = blockIdx.z;
  const int tid = threadIdx.x;
  const int p0 = nchunk * 128;
  for (int i = tid; i < 16 * CIN; i += 256) {
    int r = i / CIN, c = i % CIN;
    float sc = (c >= COUT) ? 2.0f : 1.0f;
    As[i] = woff[(size_t)(mt * 16 + r) * CIN + c] * sc;
  }
  // issue a 16-row half-chunk DMA (each half lies entirely in one source tensor)
  auto issue_half = [&](int kc, int h) {
    int row0 = kc * 32 + h * 16;
    const float* src = (row0 < COUT)
        ? feat_arm + (size_t)(b * COUT + row0) * HWp + p0
        : feat_up + (size_t)(b * COUT + (row0 - COUT)) * HWp + p0;
    tdm_load_2d(lds_off_of(Bs) + (unsigned)((kc & 1) * 32 * 128 * 4 + h * 16 * 128 * 4),
                src, 128, 16, HWp);
  };
  if (tid < 32) { issue_half(0, 0); issue_half(0, 1); }
  const int lane = tid & 31;
  const int wv = tid >> 5;
  const int n = lane & 15;
  const int hi = lane >> 4;
  const int ncol = wv * 16 + n;
  v8f acc = {};
  for (int kc = 0; kc < CIN / 32; ++kc) {
    if (tid < 32) {
      if (kc + 1 < CIN / 32) {
        issue_half(kc + 1, 0); issue_half(kc + 1, 1);
        __builtin_amdgcn_s_wait_tensorcnt(2);
      } else {
        __builtin_amdgcn_s_wait_tensorcnt(0);
      }
    }
    __syncthreads();
    const float* Bc = Bs + (kc & 1) * 32 * 128;
    #pragma unroll
    for (int kk = 0; kk < 8; ++kk) {
      int kb = kk * 4 + hi * 2;
      v2f a = *(const v2f*)(As + n * CIN + kc * 32 + kb);
      v2f bb; bb.x = Bc[kb * 128 + ncol]; bb.y = Bc[(kb + 1) * 128 + ncol];
      acc = wmma_f32x4(a, bb, acc);
    }
    __syncthreads();
  }
  // transposed store: the 8 accumulator rows are contiguous channels -> 2x float4
  const int col = p0 + ncol;
  const int m0 = mt * 16 + hi * 8;
  float* dst = offT + ((size_t)b * HWp + col) * OFFC + m0;
  float4 lo; lo.x = acc[0]; lo.y = acc[1]; lo.z = acc[2]; lo.w = acc[3];
  float4 hi4; hi4.x = acc[4]; hi4.y = acc[5]; hi4.z = acc[6]; hi4.w = acc[7];
  *(float4*)dst = lo;
  *(float4*)(dst + 4) = hi4;
}

// ---------------- kernel 6: offset BN statistics from transposed layout ----------------
__global__ void k_offstats(const float* __restrict__ offT, const float* __restrict__ gam,
                           const float* __restrict__ bet, float* __restrict__ stats) {
  const int o = blockIdx.x;                         // 0..287
  float s = 0.f, s2 = 0.f;
  for (int i = threadIdx.x; i < BB * HWp; i += 256) {
    float v = offT[(size_t)i * OFFC + o];
    s += v; s2 += v * v;
  }
  __shared__ float r1[256], r2[256];
  r1[threadIdx.x] = s; r2[threadIdx.x] = s2;
  __syncthreads();
  for (int off = 128; off > 0; off >>= 1) {
    if ((int)threadIdx.x < off) {
      r1[threadIdx.x] += r1[threadIdx.x + off];
      r2[threadIdx.x] += r2[threadIdx.x + off];
    }
    __syncthreads();
  }
  if (threadIdx.x == 0) {
    const float inv_n = 1.0f / (float)(BB * HWp);
    float mean = r1[0] * inv_n;
    float var  = r2[0] * inv_n - mean * mean;
    float sc = gam[o] * rsqrtf(var + EPSBN);
    stats[o] = sc;
    stats[OFFC + o] = bet[o] - mean * sc;
  }
}

// ---------------- kernel 7: fapn gate ----------------
__global__ void k_addweight(const float* __restrict__ feat_arm, const float* __restrict__ w,
                            const float* __restrict__ bias, float* __restrict__ aw) {
  const int idx = blockIdx.x * blockDim.x + threadIdx.x;
  if (idx >= BB * HWp) return;
  const int b = idx / HWp, p = idx % HWp;
  float s = bias[0];
  const float* base = feat_arm + (size_t)b * COUT * HWp + p;
  for (int c = 0; c < COUT; ++c) s += w[c] * base[(size_t)c * HWp];
  aw[idx] = 1.0f / (1.0f + __expf(-s));
}

// ---------------- kernel 8: deformable conv, 32-pixel tiles, dual-N WMMA ----------------
// grid: (288 tiles, 4 batches), block 288 (9 waves; wave m owns M rows [16m,16m+16))
// dynamic LDS: vt[648*32] floats (half the 1296 im2col depth at a time)
__global__ void k_deform(const float* __restrict__ fup2, const float* __restrict__ offT,
                         const float* __restrict__ stats, const float* __restrict__ dcnw,
                         const float* __restrict__ feat_arm, const float* __restrict__ aw,
                         float* __restrict__ out) {
  extern __shared__ float vt[];                     // 648 x 32
  const int b = blockIdx.y;
  const int p0 = blockIdx.x * 32;
  const int tid = threadIdx.x;
  const int lane = tid & 31;
  const int wv = tid >> 5;                          // 0..8
  const int n = lane & 15;
  const int hi = lane >> 4;
  v8f acc0 = {}, acc1 = {};
  for (int half = 0; half < 2; ++half) {
    __syncthreads();
    // ---- build: items = (32 px) x (8 groups) x (9 taps) = 2304 ----
    for (int it = tid; it < 2304; it += 288) {
      const int pn = it & 31;
      const int rest = it >> 5;                     // 0..71
      const int gl = rest / 9;
      const int k  = rest % 9;
      const int g  = half * 8 + gl;
      const int p = p0 + pn;
      const int hh = p / WW, ww = p % WW;
      const int chy = (g * KKt + k) * 2;
      float2 dxy = *(const float2*)(offT + ((size_t)b * HWp + p) * OFFC + chy);
      float2 ss = *(const float2*)(stats + chy);
      float2 sb = *(const float2*)(stats + OFFC + chy);
      float dy = dxy.x * ss.x + sb.x;
      float dx = dxy.y * ss.y + sb.y;
      float py = (float)(hh - 1 + k / 3) + dy;
      float px = (float)(ww - 1 + k % 3) + dx;
      float y0f = floorf(py), x0f = floorf(px);
      float wy = py - y0f, wx = px - x0f;
      int y0 = (int)y0f, x0 = (int)x0f;
      int y1 = y0 + 1, x1 = x0 + 1;
      float vy0 = (y0 >= 0 && y0 < HH) ? 1.f : 0.f;
      float vy1 = (y1 >= 0 && y1 < HH) ? 1.f : 0.f;
      float vx0 = (x0 >= 0 && x0 < WW) ? 1.f : 0.f;
      float vx1 = (x1 >= 0 && x1 < WW) ? 1.f : 0.f;
      int y0c = min(max(y0, 0), HH - 1), y1c = min(max(y1, 0), HH - 1);
      int x0c = min(max(x0, 0), WW - 1), x1c = min(max(x1, 0), WW - 1);
      float w00 = (1.f - wy) * (1.f - wx) * vy0 * vx0;
      float w01 = (1.f - wy) * wx * vy0 * vx1;
      float w10 = wy * (1.f - wx) * vy1 * vx0;
      float w11 = wy * wx * vy1 * vx1;
      const float* gb = fup2 + (size_t)(b * GG + g) * HWp * 16;
      float sv[9];
      #pragma unroll
      for (int c = 0; c < 9; ++c) sv[c] = 0.f;
      int ci[4] = { y0c * WW + x0c, y0c * WW + x1c, y1c * WW + x0c, y1c * WW + x1c };
      float cw[4] = { w00, w01, w10, w11 };
      #pragma unroll
      for (int q = 0; q < 4; ++q) {
        const float* cp = gb + (size_t)ci[q] * 16;
        float4 a = *(const float4*)cp;
        float4 b4 = *(const float4*)(cp + 4);
        float e = cp[8];
        float wq = cw[q];
        sv[0] += wq * a.x;  sv[1] += wq * a.y;  sv[2] += wq * a.z;  sv[3] += wq * a.w;
        sv[4] += wq * b4.x; sv[5] += wq * b4.y; sv[6] += wq * b4.z; sv[7] += wq * b4.w;
        sv[8] += wq * e;
      }
      #pragma unroll
      for (int c = 0; c < 9; ++c)
        vt[((gl * CGg + c) * KKt + k) * 32 + pn] = sv[c];
    }
    __syncthreads();
    // ---- compute: 162 K-steps of 4; A frag reused across both N tiles ----
    const float* arow = dcnw + (size_t)(wv * 16 + n) * KDC + half * 648;
    for (int ks = 0; ks < 162; ++ks) {
      int kb = ks * 4 + hi * 2;
      v2f a = *(const v2f*)(arow + kb);
      v2f b0; b0.x = vt[kb * 32 + n];      b0.y = vt[(kb + 1) * 32 + n];
      v2f b1; b1.x = vt[kb * 32 + 16 + n]; b1.y = vt[(kb + 1) * 32 + 16 + n];
      acc0 = wmma_f32x4(a, b0, acc0);
      acc1 = wmma_f32x4(a, b1, acc1);
    }
  }
  // ---- epilogue: relu + gated blend ----
  const int p = p0 + n, p2 = p0 + 16 + n;
  const float aw0 = aw[b * HWp + p];
  const float aw1 = aw[b * HWp + p2];
  #pragma unroll
  for (int r = 0; r < 8; ++r) {
    int m = wv * 16 + r + hi * 8;
    size_t i0 = (size_t)(b * COUT + m) * HWp + p;
    size_t i1 = (size_t)(b * COUT + m) * HWp + p2;
    out[i0] = aw0 * feat_arm[i0] + (1.0f - aw0) * fmaxf(acc0[r], 0.0f);
    out[i1] = aw1 * feat_arm[i1] + (1.0f - aw1) * fmaxf(acc1[r], 0.0f);
  }
}

// ---------------- host launcher ----------------
extern "C" void kernel_launch(void* const* d_in, const int* in_sizes, int n_in,
                              void* d_out, int out_size, void* d_ws, size_t ws_size,
                              hipStream_t stream) {
  const float* feat_l     = (const float*)d_in[0];
  const float* feat_s     = (const float*)d_in[1];
  const float* fsm_atten_w= (const float*)d_in[2];
  const float* fsm_bn_g   = (const float*)d_in[3];
  const float* fsm_bn_b   = (const float*)d_in[4];
  const float* fsm_conv_w = (const float*)d_in[5];
  const float* offset_w   = (const float*)d_in[6];
  const float* off_bn_g   = (const float*)d_in[7];
  const float* off_bn_b   = (const float*)d_in[8];
  const float* dcn_w      = (const float*)d_in[9];
  const float* fapn_w     = (const float*)d_in[10];
  const float* fapn_b     = (const float*)d_in[11];
  float* out = (float*)d_out;

  float* ws = (float*)d_ws;
  float* gap      = ws;                                    // 1152 (pad 1280)
  float* atten    = gap + 1280;                            // 1152 (pad 1280)
  float* feat_up  = atten + 1280;                          // 5,308,416
  float* fup2     = feat_up + (size_t)BB * COUT * HWp;     // 9,437,184
  float* feat_arm = fup2 + (size_t)BB * GG * HWp * 16;     // 5,308,416
  float* offT     = feat_arm + (size_t)BB * COUT * HWp;    // 10,616,832
  float* stats    = offT + (size_t)BB * HWp * OFFC;        // 576 (pad 1024)
  float* aw       = stats + 1024;                          // 36,864

  k_gap<<<BB * CIN, 256, 0, stream>>>(feat_l, gap);
  k_atten<<<CIN, 128, 0, stream>>>(gap, fsm_atten_w, fsm_bn_g, fsm_bn_b, atten);
  {
    int total = BB * COUT * HWp;
    k_upsample<<<(total + 255) / 256, 256, 0, stream>>>(feat_s, feat_up, fup2);
  }
  k_gemm_arm<<<dim3(HWp / 128, COUT / 16, BB), 256, 0, stream>>>(feat_l, fsm_conv_w, atten, feat_arm);
  k_gemm_off<<<dim3(HWp / 128, OFFC / 16, BB), 256, 0, stream>>>(feat_arm, feat_up, offset_w, offT);
  k_offstats<<<OFFC, 256, 0, stream>>>(offT, off_bn_g, off_bn_b, stats);
  {
    int total = BB * HWp;
    k_addweight<<<(total + 255) / 256, 256, 0, stream>>>(feat_arm, fapn_w, fapn_b, aw);
  }
  k_deform<<<dim3(HWp / 32, BB), 288, 648 * 32 * sizeof(float), stream>>>(
      fup2, offT, stats, dcn_w, feat_arm, aw, out);
}